// StationaryFactorizedNoise_39522289058115
// MI455X (gfx1250) — compile-verified
//
#include <hip/hip_runtime.h>

typedef __attribute__((ext_vector_type(2))) float v2f;
typedef __attribute__((ext_vector_type(8))) float v8f;

// Problem constants (match setup_inputs)
constexpr int SIZE  = 4;
constexpr int C     = 384;
constexpr int SIMT  = 65536;
constexpr int KT    = 121;
constexpr int BS    = 1024;                 // FFT block
constexpr int STEP  = BS - KT + 1;          // 904
constexpr int T     = SIMT + KT - 1;        // 65656
constexpr int NBLK  = (T + STEP - 1)/STEP;  // 73
constexpr int NROWS = SIZE * C;             // 1536
constexpr int NCOLS = NROWS * NBLK;         // 112128
constexpr int OFF   = KT - 1;               // 120
constexpr int K1    = STEP / 4;             // 226 K-steps, stage 1
constexpr int K2    = C / 4;                // 96  K-steps, stage 2

// workspace layout (floats)
constexpr size_t WS_H    = 0;
constexpr size_t WS_S    = 1024;
constexpr size_t WS_FILT = 1024 + (size_t)C*C;   // 148480
constexpr size_t FILT_N  = (size_t)NROWS * SIMT; // 100,663,296

// ---------------- prep: h = irfft(kernel_fft, 1024), backward norm ----------------
__global__ void prep_h(const float* __restrict__ f2, float* __restrict__ h) {
  int t = blockIdx.x * blockDim.x + threadIdx.x;
  if (t >= BS) return;
  float s = 0.f;
  const float w = 6.135923151e-3f; // 2*pi/1024
  for (int k = 1; k < BS/2; ++k) {
    int m = (k * t) & (BS - 1);    // exact phase via integer mod
    s += f2[k] * cosf((float)m * w);
  }
  float nyq = f2[BS/2] * ((t & 1) ? -1.f : 1.f);
  h[t] = (f2[0] + 2.f * s + nyq) * (1.f / (float)BS);
}

// ---------------- prep: S[c,d] = std[c] * Vt[c,d] ----------------
__global__ void prep_S(const float* __restrict__ stdv, const float* __restrict__ vt,
                       float* __restrict__ S) {
  int e = blockIdx.x * blockDim.x + threadIdx.x;
  if (e < C * C) S[e] = stdv[e / C] * vt[e];
}

// ---------------- zero the filt intermediate ----------------
__global__ void zero_filt(float4* __restrict__ p, size_t n4) {
  size_t i = (size_t)blockIdx.x * blockDim.x + threadIdx.x;
  size_t stride = (size_t)gridDim.x * blockDim.x;
  float4 z = {0.f, 0.f, 0.f, 0.f};
  for (; i < n4; i += stride) p[i] = z;
}

// ---------------- stage 1: circular-conv GEMM + overlap-add ----------------
// One block = 16 (row,block) columns, all 64 j-tiles. 8 waves x 8 tiles each.
// filt[t] gets exactly one contribution unless j in [0,120) U [904,1024)
// (the block_size-step = 120 sample seams) -> plain stores in the exclusive
// region, global_atomic_add_f32 only at the seams.
__global__ __launch_bounds__(256) void conv_oa(const float* __restrict__ noise,
                                               const float* __restrict__ hws,
                                               float* __restrict__ filt) {
  __shared__ float hs[BS];            // 4 KB
  __shared__ float Xs[16 * 908];      // 58,112 B, [col*908 + i]
  __shared__ int   colR[16], colB[16];

  const int tid = threadIdx.x;
  const int nb0 = blockIdx.x * 16;

  for (int i = tid; i < BS; i += 256) hs[i] = hws[i];
  if (tid < 16) { int col = nb0 + tid; colR[tid] = col / NBLK; colB[tid] = col % NBLK; }

  { // frames -> LDS, 16 threads per column, coalesced in i
    int col = tid >> 4;
    int cg  = nb0 + col;
    int r   = cg / NBLK, b = cg % NBLK;
    size_t base = (size_t)r * T + (size_t)b * STEP;
    for (int i = (tid & 15); i < 908; i += 16) {
      float v = 0.f;
      if (i < STEP && (b * STEP + i) < T) v = noise[base + i];
      Xs[col * 908 + i] = v;
    }
  }
  __syncthreads();

  const int wave = tid >> 5, lane = tid & 31;
  const int lm = lane & 15, lh = lane >> 4;
  const int r = colR[lm], b = colB[lm];
  float* frow = filt + (size_t)r * SIMT;

  for (int jt = 0; jt < 8; ++jt) {
    const int j0 = (wave * 8 + jt) * 16;
    v8f acc = {0.f,0.f,0.f,0.f,0.f,0.f,0.f,0.f};
    for (int ks = 0; ks < K1; ++ks) {
      const int k = ks * 4 + lh * 2;
      v2f a;
      a.x = hs[(j0 + lm - k    ) & (BS - 1)];   // H[j, k]   = h[(j-k) mod 1024]
      a.y = hs[(j0 + lm - k - 1) & (BS - 1)];   // H[j, k+1]
      const v2f bb = *(const v2f*)&Xs[lm * 908 + k];
      acc = __builtin_amdgcn_wmma_f32_16x16x4_f32(
          false, a, false, bb, (short)0, acc, false, false);
    }
    // overlap-add into filt: t = b*step + j - 120
    const int jbase = j0 + lh * 8;
    const int tbase = b * STEP - OFF + jbase;
    for (int v = 0; v < 8; ++v) {
      const int j = jbase + v;          // uniform per half-wave
      const int t = tbase + v;
      if (t >= 0 && t < SIMT) {
        if (j >= OFF && j < STEP) {
          frow[t] = acc[v];             // exclusive region: single contributor
        } else {
          __hip_atomic_fetch_add(&frow[t], acc[v],
                                 __ATOMIC_RELAXED, __HIP_MEMORY_SCOPE_AGENT);
        }
      }
    }
  }
}

// ---------------- stage 2: out[n,t,d] = sum_c filt[n,c,t] * S[c,d] ----------------
// One block = one d-tile (16) x 8 consecutive t-tiles (one per wave), fixed n.
__global__ __launch_bounds__(256) void spatial_mix(const float* __restrict__ filt,
                                                   const float* __restrict__ S,
                                                   float* __restrict__ out) {
  __shared__ float Ss[C * 16];        // 24 KB, [c*16 + dcol]

  const int bid = blockIdx.x;
  const int dt  = bid % (C / 16);
  const int tmp = bid / (C / 16);
  const int ttg = tmp % (SIMT / 16 / 8);
  const int n   = tmp / (SIMT / 16 / 8);
  const int d0  = dt * 16;
  const int tid = threadIdx.x;

  for (int e = tid; e < C * 16; e += 256) {
    int c = e >> 4, dc = e & 15;
    Ss[e] = S[c * C + d0 + dc];
  }
  __syncthreads();

  const int wave = tid >> 5, lane = tid & 31;
  const int lm = lane & 15, lh = lane >> 4;
  const int t0 = (ttg * 8 + wave) * 16;

  const float* Abase = filt + (size_t)(n * C) * SIMT + t0 + lm;

  v8f acc = {0.f,0.f,0.f,0.f,0.f,0.f,0.f,0.f};
  for (int ks = 0; ks < K2; ++ks) {
    const int c = ks * 4 + lh * 2;
    // prefetch the A stream a few K-rows ahead (global_prefetch_b8;
    // speculative, OOB at the tail is silently dropped)
    __builtin_prefetch(Abase + (size_t)(c + 16) * SIMT, 0, 1);
    v2f a;
    a.x = Abase[(size_t)c       * SIMT];
    a.y = Abase[(size_t)(c + 1) * SIMT];
    v2f bb;
    bb.x = Ss[c * 16 + lm];
    bb.y = Ss[(c + 1) * 16 + lm];
    acc = __builtin_amdgcn_wmma_f32_16x16x4_f32(
        false, a, false, bb, (short)0, acc, false, false);
  }
  // out is write-once, never re-read: nontemporal stores keep the 192MB L2
  // free for the heavily reused filt A-tiles and the resident S matrix.
  for (int v = 0; v < 8; ++v) {
    int t = t0 + v + lh * 8;
    __builtin_nontemporal_store(acc[v],
        &out[((size_t)(n * SIMT + t)) * C + d0 + lm]);
  }
}

extern "C" void kernel_launch(void* const* d_in, const int* in_sizes, int n_in,
                              void* d_out, int out_size, void* d_ws, size_t ws_size,
                              hipStream_t stream) {
  (void)in_sizes; (void)n_in; (void)out_size; (void)ws_size;
  const float* noise   = (const float*)d_in[0];
  const float* stdv    = (const float*)d_in[1];
  const float* vt      = (const float*)d_in[2];
  const float* kfft    = (const float*)d_in[3];
  float* w    = (float*)d_ws;
  float* h    = w + WS_H;
  float* S    = w + WS_S;
  float* filt = w + WS_FILT;
  float* out  = (float*)d_out;

  prep_h<<<(BS + 255) / 256, 256, 0, stream>>>(kfft, h);
  prep_S<<<(C * C + 255) / 256, 256, 0, stream>>>(stdv, vt, S);
  zero_filt<<<2048, 256, 0, stream>>>((float4*)filt, FILT_N / 4);
  conv_oa<<<NCOLS / 16, 256, 0, stream>>>(noise, h, filt);
  spatial_mix<<<SIZE * (SIMT / 16 / 8) * (C / 16), 256, 0, stream>>>(filt, S, out);
}